// GATLinNet_18116172055283
// MI455X (gfx1250) — compile-verified
//
#include <hip/hip_runtime.h>
#include <math.h>

#define Nn 50000
#define Ee 800000
#define EtT (Ee + Nn)
#define D1 256
#define NEG_SLOPE 0.2f

typedef float v2f __attribute__((ext_vector_type(2)));
typedef float v8f __attribute__((ext_vector_type(8)));

// ---------------------------------------------------------------------------
// float atomic max via signed-max / unsigned-min bit trick (init = -inf)
// ---------------------------------------------------------------------------
__device__ __forceinline__ void atomicMaxF(float* addr, float v) {
    if (v >= 0.0f) atomicMax((int*)addr, __float_as_int(v));
    else           atomicMin((unsigned int*)addr, __float_as_uint(v));
}

__device__ __forceinline__ float lrelu(float x) { return x > 0.0f ? x : NEG_SLOPE * x; }

// ---------------------------------------------------------------------------
// fill
// ---------------------------------------------------------------------------
__global__ void fill_kernel(float* __restrict__ p, float v, int n) {
    int i = blockIdx.x * blockDim.x + threadIdx.x;
    if (i < n) p[i] = v;
}

// ---------------------------------------------------------------------------
// Repack W[K x 256] into K-pair-interleaved float2 so a WMMA B-fragment
// (rows kk, kk+1 at one column) is a single global_load_b64.
// Wp[p*256 + n] = { W[2p, n], W[2p+1, n] }
// ---------------------------------------------------------------------------
__global__ void pack_w(const float* __restrict__ W, float2* __restrict__ Wp, int K) {
    int i = blockIdx.x * blockDim.x + threadIdx.x;
    if (i >= (K >> 1) * D1) return;
    int p = i >> 8, n = i & 255;
    Wp[i] = make_float2(W[(size_t)(2 * p) * D1 + n], W[(size_t)(2 * p + 1) * D1 + n]);
}

// ---------------------------------------------------------------------------
// Fused dual GEMM: Cc = A@Wc,  Cl = A@Wl + bias0 + bias1   (both M x 256)
// fp32 WMMA 16x16x4. Block = 256 thr = 8 waves, 32 rows/block.
// A block staged global->LDS with gfx1250 async-copy (ASYNCcnt) when full;
// guarded scalar fallback for the ragged tail block.
// ---------------------------------------------------------------------------
__global__ __launch_bounds__(256) void gemm_dual(
        const float* __restrict__ A,
        const float2* __restrict__ WpC, const float2* __restrict__ WpL,
        float* __restrict__ Cc, float* __restrict__ Cl,
        int M, int K,
        const float* __restrict__ bias0, const float* __restrict__ bias1) {
    __shared__ float As[32 * 260];                 // pad 260: conflict-free column reads
    const int tid     = threadIdx.x;
    const int rowBase = blockIdx.x * 32;
    const int K4      = K >> 2;                    // 16B chunks per row

    if (rowBase + 32 <= M) {
        // async DMA: 16 bytes per issue, direct global -> LDS, no VGPR bounce
        for (int i = tid; i < 32 * K4; i += 256) {
            int r = i / K4, c4 = i - r * K4;
            unsigned int lds = (unsigned int)(size_t)&As[r * 260 + c4 * 4];
            const float* g = A + (size_t)(rowBase + r) * K + c4 * 4;
            asm volatile("global_load_async_to_lds_b128 %0, %1, off"
                         :: "v"(lds), "v"(g) : "memory");
        }
    } else {
        for (int i = tid; i < 32 * K; i += 256) {
            int r = i / K, c = i - r * K;
            int gr = rowBase + r;
            As[r * 260 + c] = (gr < M) ? A[(size_t)gr * K + c] : 0.0f;
        }
    }
    asm volatile("s_wait_asynccnt 0" ::: "memory");
    __syncthreads();

    const int lane = tid & 31;
    const int w    = tid >> 5;
    const int msub = w >> 2;            // 0..1   -> rows msub*16..+15
    const int nq   = w & 3;             // 0..3   -> cols nq*64..+63
    const int hi   = lane >> 4;         // half-wave: K pair select
    const int l15  = lane & 15;
    const int rowLocal = msub * 16 + l15;
    const int colBase  = nq * 64 + l15;

    v8f zero = {0.f,0.f,0.f,0.f,0.f,0.f,0.f,0.f};
    v8f accC[4] = {zero, zero, zero, zero};
    v8f accL[4] = {zero, zero, zero, zero};

    for (int k = 0; k < K; k += 4) {
        v2f a;
        a.x = As[rowLocal * 260 + k + 2 * hi];
        a.y = As[rowLocal * 260 + k + 2 * hi + 1];
        const int p = (k >> 1) + hi;
        const float2* rc = WpC + (size_t)p * D1 + colBase;
        const float2* rl = WpL + (size_t)p * D1 + colBase;
#pragma unroll
        for (int t = 0; t < 4; ++t) {
            float2 bc = rc[t * 16];
            v2f b; b.x = bc.x; b.y = bc.y;
            accC[t] = __builtin_amdgcn_wmma_f32_16x16x4_f32(
                          false, a, false, b, (short)0, accC[t], false, false);
            float2 bl = rl[t * 16];
            b.x = bl.x; b.y = bl.y;
            accL[t] = __builtin_amdgcn_wmma_f32_16x16x4_f32(
                          false, a, false, b, (short)0, accL[t], false, false);
        }
    }

    // C/D layout: VGPR i -> row i (lanes 0-15), row i+8 (lanes 16-31)
#pragma unroll
    for (int t = 0; t < 4; ++t) {
#pragma unroll
        for (int i = 0; i < 8; ++i) {
            int row = rowBase + msub * 16 + hi * 8 + i;
            if (row < M) {
                int col = nq * 64 + t * 16 + l15;
                Cc[(size_t)row * D1 + col] = accC[t][i];
                Cl[(size_t)row * D1 + col] = accL[t][i] + bias0[col] + bias1[col];
            }
        }
    }
}

// ---------------------------------------------------------------------------
// a_s[n,h] = sum_c xl[n,h,c]*att_src[h,c] ; a_d likewise  (C = 64, H = 4)
// ---------------------------------------------------------------------------
__global__ void attn_scores(const float* __restrict__ xl,
                            const float* __restrict__ cas,
                            const float* __restrict__ cad,
                            float* __restrict__ a_s, float* __restrict__ a_d) {
    int idx = blockIdx.x * blockDim.x + threadIdx.x;     // n*4 threads
    if (idx >= Nn * 4) return;
    int node = idx >> 2, h = idx & 3;
    const float4* xr = (const float4*)(xl + (size_t)node * D1 + h * 64);
    const float4* sr = (const float4*)(cas + h * 64);
    const float4* dr = (const float4*)(cad + h * 64);
    float ss = 0.f, dd = 0.f;
#pragma unroll
    for (int j = 0; j < 16; ++j) {
        float4 v = xr[j], s = sr[j], d = dr[j];
        ss += v.x*s.x + v.y*s.y + v.z*s.z + v.w*s.w;
        dd += v.x*d.x + v.y*d.y + v.z*d.z + v.w*d.w;
    }
    a_s[idx] = ss;  a_d[idx] = dd;
}

// layer 2: C = 1 -> score is a single product per head
__global__ void attn_c1(const float* __restrict__ xl2,
                        const float* __restrict__ cas2, const float* __restrict__ cad2,
                        float* __restrict__ a_s, float* __restrict__ a_d) {
    int n = blockIdx.x * blockDim.x + threadIdx.x;
    if (n >= Nn) return;
    float4 v = *(const float4*)(xl2 + (size_t)n * 4);
    float4 s, d;
    s.x = v.x*cas2[0]; s.y = v.y*cas2[1]; s.z = v.z*cas2[2]; s.w = v.w*cas2[3];
    d.x = v.x*cad2[0]; d.y = v.y*cad2[1]; d.z = v.z*cad2[2]; d.w = v.w*cad2[3];
    *(float4*)(a_s + (size_t)n * 4) = s;
    *(float4*)(a_d + (size_t)n * 4) = d;
}

// ---------------------------------------------------------------------------
// edge softmax passes (thread per edge; self-loops appended implicitly)
// ---------------------------------------------------------------------------
__device__ __forceinline__ void edge_sd(const int* __restrict__ ei, int e, int& s, int& d) {
    if (e < Ee) { s = ei[e]; d = ei[Ee + e]; }
    else        { s = d = e - Ee; }
}

__global__ void edge_max(const int* __restrict__ ei,
                         const float* __restrict__ a_s, const float* __restrict__ a_d,
                         float* __restrict__ emax) {
    int e = blockIdx.x * blockDim.x + threadIdx.x;
    if (e >= EtT) return;
    int s, d; edge_sd(ei, e, s, d);
    float4 as = *(const float4*)(a_s + (size_t)s * 4);
    float4 ad = *(const float4*)(a_d + (size_t)d * 4);
    atomicMaxF(&emax[d*4+0], lrelu(as.x + ad.x));
    atomicMaxF(&emax[d*4+1], lrelu(as.y + ad.y));
    atomicMaxF(&emax[d*4+2], lrelu(as.z + ad.z));
    atomicMaxF(&emax[d*4+3], lrelu(as.w + ad.w));
}

__global__ void edge_denom(const int* __restrict__ ei,
                           const float* __restrict__ a_s, const float* __restrict__ a_d,
                           const float* __restrict__ emax, float* __restrict__ denom) {
    int e = blockIdx.x * blockDim.x + threadIdx.x;
    if (e >= EtT) return;
    int s, d; edge_sd(ei, e, s, d);
    float4 as = *(const float4*)(a_s + (size_t)s * 4);
    float4 ad = *(const float4*)(a_d + (size_t)d * 4);
    float4 mx = *(const float4*)(emax + (size_t)d * 4);
    atomicAdd(&denom[d*4+0], __expf(lrelu(as.x + ad.x) - mx.x));
    atomicAdd(&denom[d*4+1], __expf(lrelu(as.y + ad.y) - mx.y));
    atomicAdd(&denom[d*4+2], __expf(lrelu(as.z + ad.z) - mx.z));
    atomicAdd(&denom[d*4+3], __expf(lrelu(as.w + ad.w) - mx.w));
}

// message scatter, C=64: one wave32 per edge, 8 channels per lane
__global__ __launch_bounds__(256) void edge_scatter(
        const int* __restrict__ ei,
        const float* __restrict__ a_s, const float* __restrict__ a_d,
        const float* __restrict__ emax, const float* __restrict__ denom,
        const float* __restrict__ xl, float* __restrict__ acc) {
    int gw   = (blockIdx.x * 256 + threadIdx.x) >> 5;
    int lane = threadIdx.x & 31;
    if (gw >= EtT) return;
    int s, d; edge_sd(ei, gw, s, d);
    const float* xs = xl + (size_t)s * D1;
    float*       ac = acc + (size_t)d * D1;
    __builtin_prefetch(xs + lane, 0, 1);           // global_prefetch_b8: gather row
    __builtin_prefetch(ac + lane, 1, 1);           // scatter row
    float4 as = *(const float4*)(a_s + (size_t)s * 4);
    float4 ad = *(const float4*)(a_d + (size_t)d * 4);
    float4 mx = *(const float4*)(emax + (size_t)d * 4);
    float4 dn = *(const float4*)(denom + (size_t)d * 4);
    float al0 = __expf(lrelu(as.x + ad.x) - mx.x) / (dn.x + 1e-16f);
    float al1 = __expf(lrelu(as.y + ad.y) - mx.y) / (dn.y + 1e-16f);
    float al2 = __expf(lrelu(as.z + ad.z) - mx.z) / (dn.z + 1e-16f);
    float al3 = __expf(lrelu(as.w + ad.w) - mx.w) / (dn.w + 1e-16f);
#pragma unroll
    for (int j = 0; j < 8; ++j) {
        int c = lane + 32 * j;
        int h = c >> 6;
        float a = (h == 0) ? al0 : (h == 1) ? al1 : (h == 2) ? al2 : al3;
        atomicAdd(&ac[c], a * xs[c]);
    }
}

// message scatter, C=1 (layer 2): thread per edge
__global__ void edge_scatter2(const int* __restrict__ ei,
        const float* __restrict__ a_s, const float* __restrict__ a_d,
        const float* __restrict__ emax, const float* __restrict__ denom,
        const float* __restrict__ xl2, float* __restrict__ agg2) {
    int e = blockIdx.x * blockDim.x + threadIdx.x;
    if (e >= EtT) return;
    int s, d; edge_sd(ei, e, s, d);
    float4 as = *(const float4*)(a_s + (size_t)s * 4);
    float4 ad = *(const float4*)(a_d + (size_t)d * 4);
    float4 mx = *(const float4*)(emax + (size_t)d * 4);
    float4 dn = *(const float4*)(denom + (size_t)d * 4);
    float4 xv = *(const float4*)(xl2 + (size_t)s * 4);
    atomicAdd(&agg2[d*4+0], __expf(lrelu(as.x + ad.x) - mx.x) / (dn.x + 1e-16f) * xv.x);
    atomicAdd(&agg2[d*4+1], __expf(lrelu(as.y + ad.y) - mx.y) / (dn.y + 1e-16f) * xv.y);
    atomicAdd(&agg2[d*4+2], __expf(lrelu(as.z + ad.z) - mx.z) / (dn.z + 1e-16f) * xv.z);
    atomicAdd(&agg2[d*4+3], __expf(lrelu(as.w + ad.w) - mx.w) / (dn.w + 1e-16f) * xv.w);
}

// in-place ELU (acc already holds gat_agg + cb + lin + lb)
__global__ void elu_kernel(float* __restrict__ p, int n) {
    int i = blockIdx.x * blockDim.x + threadIdx.x;
    if (i >= n) return;
    float v = p[i];
    p[i] = v > 0.0f ? v : (__expf(v) - 1.0f);
}

// layer-2 GEMV: xl2[n,0..3] = h[n,:]@cW2 ; lin2[n] = h[n,:]@lW2 + lb2
__global__ __launch_bounds__(256) void small_gemv(
        const float* __restrict__ h, const float* __restrict__ cW2,
        const float* __restrict__ lW2, const float* __restrict__ lb2,
        float* __restrict__ xl2, float* __restrict__ lin2) {
    int gw   = (blockIdx.x * 256 + threadIdx.x) >> 5;
    int lane = threadIdx.x & 31;
    if (gw >= Nn) return;
    const float* hr = h + (size_t)gw * D1;
    float a0 = 0.f, a1 = 0.f, a2 = 0.f, a3 = 0.f, a4 = 0.f;
#pragma unroll
    for (int j = 0; j < 8; ++j) {
        int k = lane + 32 * j;
        float hv = hr[k];
        const float* wr = cW2 + (size_t)k * 4;
        a0 += hv * wr[0]; a1 += hv * wr[1]; a2 += hv * wr[2]; a3 += hv * wr[3];
        a4 += hv * lW2[k];
    }
#pragma unroll
    for (int off = 16; off > 0; off >>= 1) {
        a0 += __shfl_xor(a0, off, 32);
        a1 += __shfl_xor(a1, off, 32);
        a2 += __shfl_xor(a2, off, 32);
        a3 += __shfl_xor(a3, off, 32);
        a4 += __shfl_xor(a4, off, 32);
    }
    if (lane == 0) {
        xl2[(size_t)gw * 4 + 0] = a0;
        xl2[(size_t)gw * 4 + 1] = a1;
        xl2[(size_t)gw * 4 + 2] = a2;
        xl2[(size_t)gw * 4 + 3] = a3;
        lin2[gw] = a4 + lb2[0];
    }
}

// out[n] = mean_h agg2[n,h] + cb2 + lin2[n]
__global__ void final_out(const float* __restrict__ agg2, const float* __restrict__ lin2,
                          const float* __restrict__ cb2, float* __restrict__ out) {
    int n = blockIdx.x * blockDim.x + threadIdx.x;
    if (n >= Nn) return;
    float4 g = *(const float4*)(agg2 + (size_t)n * 4);
    out[n] = 0.25f * (g.x + g.y + g.z + g.w) + cb2[0] + lin2[n];
}

// ---------------------------------------------------------------------------
extern "C" void kernel_launch(void* const* d_in, const int* in_sizes, int n_in,
                              void* d_out, int out_size, void* d_ws, size_t ws_size,
                              hipStream_t stream) {
    const float* x   = (const float*)d_in[0];
    const int*   ei  = (const int*)d_in[1];
    const float* cW0 = (const float*)d_in[3];
    const float* cas0= (const float*)d_in[4];
    const float* cad0= (const float*)d_in[5];
    const float* cb0 = (const float*)d_in[6];
    const float* lW0 = (const float*)d_in[7];
    const float* lb0 = (const float*)d_in[8];
    const float* cW1 = (const float*)d_in[9];
    const float* cas1= (const float*)d_in[10];
    const float* cad1= (const float*)d_in[11];
    const float* cb1 = (const float*)d_in[12];
    const float* lW1 = (const float*)d_in[13];
    const float* lb1 = (const float*)d_in[14];
    const float* cW2 = (const float*)d_in[15];
    const float* cas2= (const float*)d_in[16];
    const float* cad2= (const float*)d_in[17];
    const float* cb2 = (const float*)d_in[18];
    const float* lW2 = (const float*)d_in[19];
    const float* lb2 = (const float*)d_in[20];
    float* out = (float*)d_out;

    // workspace arena (floats; all offsets 16B-aligned)
    float* f = (float*)d_ws;
    float* h0   = f; f += (size_t)Nn * D1;
    float* h1   = f; f += (size_t)Nn * D1;
    float* xl   = f; f += (size_t)Nn * D1;
    float* a_s  = f; f += (size_t)Nn * 4;
    float* a_d  = f; f += (size_t)Nn * 4;
    float* emax = f; f += (size_t)Nn * 4;
    float* deno = f; f += (size_t)Nn * 4;
    float* xl2  = f; f += (size_t)Nn * 4;
    float* agg2 = f; f += (size_t)Nn * 4;
    float* lin2 = f; f += (size_t)Nn;
    float2* wpC0 = (float2*)f; f += (size_t)64  * D1 * 2;   // K=128 -> 64 pairs
    float2* wpL0 = (float2*)f; f += (size_t)64  * D1 * 2;
    float2* wpC1 = (float2*)f; f += (size_t)128 * D1 * 2;   // K=256 -> 128 pairs
    float2* wpL1 = (float2*)f; f += (size_t)128 * D1 * 2;

    const int gemmBlocks = (Nn + 31) / 32;
    const int edgeBlocks = (EtT + 255) / 256;
    const int scatBlocks = (EtT * 32 + 255) / 256;   // wave per edge
    const int nh4Blocks  = (Nn * 4 + 255) / 256;
    const int nBlocks    = (Nn + 255) / 256;
    const int featBlocks = (Nn * D1 + 255) / 256;
    const int gemvBlocks = (Nn * 32 + 255) / 256;
    const int pack0Blks  = (64  * D1 + 255) / 256;
    const int pack1Blks  = (128 * D1 + 255) / 256;

    // one-shot weight repacks (same stream -> ordered before GEMMs)
    pack_w<<<pack0Blks, 256, 0, stream>>>(cW0, wpC0, 128);
    pack_w<<<pack0Blks, 256, 0, stream>>>(lW0, wpL0, 128);
    pack_w<<<pack1Blks, 256, 0, stream>>>(cW1, wpC1, 256);
    pack_w<<<pack1Blks, 256, 0, stream>>>(lW1, wpL1, 256);

    // ---------------- layer 0: input x (K=128), output h0 ----------------
    gemm_dual<<<gemmBlocks, 256, 0, stream>>>(x, wpC0, wpL0, xl, h0, Nn, 128, lb0, cb0);
    attn_scores<<<nh4Blocks, 256, 0, stream>>>(xl, cas0, cad0, a_s, a_d);
    fill_kernel<<<nh4Blocks, 256, 0, stream>>>(emax, -INFINITY, Nn * 4);
    fill_kernel<<<nh4Blocks, 256, 0, stream>>>(deno, 0.0f, Nn * 4);
    edge_max  <<<edgeBlocks, 256, 0, stream>>>(ei, a_s, a_d, emax);
    edge_denom<<<edgeBlocks, 256, 0, stream>>>(ei, a_s, a_d, emax, deno);
    edge_scatter<<<scatBlocks, 256, 0, stream>>>(ei, a_s, a_d, emax, deno, xl, h0);
    elu_kernel<<<featBlocks, 256, 0, stream>>>(h0, Nn * D1);

    // ---------------- layer 1: input h0 (K=256), output h1 ----------------
    gemm_dual<<<gemmBlocks, 256, 0, stream>>>(h0, wpC1, wpL1, xl, h1, Nn, 256, lb1, cb1);
    attn_scores<<<nh4Blocks, 256, 0, stream>>>(xl, cas1, cad1, a_s, a_d);
    fill_kernel<<<nh4Blocks, 256, 0, stream>>>(emax, -INFINITY, Nn * 4);
    fill_kernel<<<nh4Blocks, 256, 0, stream>>>(deno, 0.0f, Nn * 4);
    edge_max  <<<edgeBlocks, 256, 0, stream>>>(ei, a_s, a_d, emax);
    edge_denom<<<edgeBlocks, 256, 0, stream>>>(ei, a_s, a_d, emax, deno);
    edge_scatter<<<scatBlocks, 256, 0, stream>>>(ei, a_s, a_d, emax, deno, xl, h1);
    elu_kernel<<<featBlocks, 256, 0, stream>>>(h1, Nn * D1);

    // ---------------- layer 2: input h1, C=1 per head, mean ----------------
    small_gemv<<<gemvBlocks, 256, 0, stream>>>(h1, cW2, lW2, lb2, xl2, lin2);
    attn_c1<<<nBlocks, 256, 0, stream>>>(xl2, cas2, cad2, a_s, a_d);
    fill_kernel<<<nh4Blocks, 256, 0, stream>>>(emax, -INFINITY, Nn * 4);
    fill_kernel<<<nh4Blocks, 256, 0, stream>>>(deno, 0.0f, Nn * 4);
    fill_kernel<<<nh4Blocks, 256, 0, stream>>>(agg2, 0.0f, Nn * 4);
    edge_max  <<<edgeBlocks, 256, 0, stream>>>(ei, a_s, a_d, emax);
    edge_denom<<<edgeBlocks, 256, 0, stream>>>(ei, a_s, a_d, emax, deno);
    edge_scatter2<<<edgeBlocks, 256, 0, stream>>>(ei, a_s, a_d, emax, deno, xl2, agg2);
    final_out<<<nBlocks, 256, 0, stream>>>(agg2, lin2, cb2, out);
}